// ResidualGCN_44633300140136
// MI455X (gfx1250) — compile-verified
//
#include <hip/hip_runtime.h>

#define NG 64            // graphs
#define GN_EPS 1e-5f
#define USE_TDM 1

typedef float v2f __attribute__((ext_vector_type(2)));
typedef float v8f __attribute__((ext_vector_type(8)));
typedef unsigned int u32x4 __attribute__((ext_vector_type(4)));
typedef unsigned int u32x8 __attribute__((ext_vector_type(8)));

// ---------------------------------------------------------------------------
// Tensor Data Mover: 2-D tile load Global -> LDS (CDNA5 ISA/§8).
// Descriptor built in SGPRs (all inputs are wave-uniform); groups 2/3 zeroed
// (2-D tile: tile_dim2/3 unused). Tensor dims measured from the tile start.
// ---------------------------------------------------------------------------
__device__ __forceinline__ void tdm_load_2d(unsigned long long gaddr,  // byte addr of tile start
                                            unsigned int lds_addr,    // byte offset in LDS
                                            unsigned int td0, unsigned int td1,       // tensor dims (elems)
                                            unsigned int tile0, unsigned int tile1,   // tile dims (elems)
                                            unsigned long long stride0) {             // dim0 stride (elems)
    u32x4 g0;
    g0[0] = 1u;                                               // count=1, user mode
    g0[1] = lds_addr;
    g0[2] = (unsigned int)gaddr;                              // global_addr[31:0]
    g0[3] = ((unsigned int)(gaddr >> 32) & 0x01FFFFFFu)       // global_addr[56:32]
            | (2u << 30);                                     // type=2 ("image")
    u32x8 g1;
    g1[0] = 2u << 16;                                         // data_size=4B; wg_mask=0
    g1[1] = (td0 & 0xFFFFu) << 16;                            // tensor_dim0[15:0] -> bits 63:48
    g1[2] = ((td0 >> 16) & 0xFFFFu) | ((td1 & 0xFFFFu) << 16);
    g1[3] = ((td1 >> 16) & 0xFFFFu) | ((tile0 & 0xFFFFu) << 16);
    g1[4] = tile1 & 0xFFFFu;                                  // tile_dim1; tile_dim2=0
    g1[5] = (unsigned int)stride0;                            // tensor_dim0_stride[31:0]
    g1[6] = (unsigned int)((stride0 >> 32) & 0xFFFFu);        // stride[47:32]; dim1_stride=0
    g1[7] = 0u;
    u32x4 gz = {0u, 0u, 0u, 0u};
    asm volatile("tensor_load_to_lds %0, %1, %2, %3"
                 :
                 : "s"(g0), "s"(g1), "s"(gz), "s"(gz)
                 : "memory");
}

// ---------------------------------------------------------------------------
// Small utility kernels
// ---------------------------------------------------------------------------
__global__ void k_fill(float* p, float v, int n) {
    int i = blockIdx.x * blockDim.x + threadIdx.x;
    if (i < n) p[i] = v;
}

__global__ void k_deg_edges(const int* __restrict__ dst, float* deg, int E) {
    int e = blockIdx.x * blockDim.x + threadIdx.x;
    if (e < E) atomicAdd(&deg[dst[e]], 1.0f);
}

__global__ void k_rsqrt_ip(float* p, int n) {
    int i = blockIdx.x * blockDim.x + threadIdx.x;
    if (i < n) p[i] = rsqrtf(p[i]);
}

__global__ void k_counts(const int* __restrict__ batch, float* counts, int n) {
    int i = blockIdx.x * blockDim.x + threadIdx.x;
    if (i < n) atomicAdd(&counts[batch[i]], 1.0f);
}

// ---------------------------------------------------------------------------
// GEMM: out[N x Cout] = A[N x K] @ B[K x Cout] via V_WMMA_F32_16X16X4_F32.
// Block = 256 threads (8 waves). Block tile: 128 rows x 16 cols.
// TDM stages A(128xK) and B(Kx16) into LDS; WMMA consumes from LDS.
// A-frag layout: lane m=lane&15, K pair {0,1}(lanes0-15)/{2,3}(lanes16-31).
// D layout: VGPR r -> row base+r (+8 for upper half-wave), col = lane&15.
// ---------------------------------------------------------------------------
template <int K, int BM>
__global__ __launch_bounds__(256) void k_gemm_tdm(const float* __restrict__ A,
                                                  const float* __restrict__ B,
                                                  float* __restrict__ out,
                                                  int Cout, int N) {
    __shared__ float As[BM * K];
    __shared__ float Bs[K * 16];

    const int m0 = blockIdx.x * BM;
    const int tn = blockIdx.y;
    const int rows = min(BM, N - m0);

#if USE_TDM
    if (threadIdx.x < 32) {   // wave 0 issues both DMAs (TDM ignores EXEC)
        unsigned long long gA = (unsigned long long)(uintptr_t)(A + (size_t)m0 * K);
        unsigned long long gB = (unsigned long long)(uintptr_t)(B + (size_t)tn * 16);
        unsigned int ldsA = (unsigned int)(uintptr_t)&As[0];   // flat low 32b == LDS offset
        unsigned int ldsB = (unsigned int)(uintptr_t)&Bs[0];
        // A tile: rows x K, row length K, remaining rows N-m0
        tdm_load_2d(gA, ldsA, (unsigned)K, (unsigned)(N - m0),
                    (unsigned)K, (unsigned)rows, (unsigned long long)K);
        // B tile: K x 16 at col tn*16; remaining row length Cout - tn*16
        tdm_load_2d(gB, ldsB, (unsigned)(Cout - tn * 16), (unsigned)K,
                    16u, (unsigned)K, (unsigned long long)Cout);
        if (m0 + BM < N)      // warm L2 for the next M-strip
            __builtin_prefetch(A + (size_t)(m0 + BM) * K, 0, 0);
        __builtin_amdgcn_s_wait_tensorcnt(0);
    }
    __syncthreads();
#else
    for (int t = threadIdx.x; t < rows * K; t += 256)
        As[(t / K) * K + (t % K)] = A[(size_t)(m0 + t / K) * K + (t % K)];
    for (int t = threadIdx.x; t < K * 16; t += 256)
        Bs[t] = B[(size_t)(t / 16) * Cout + tn * 16 + (t % 16)];
    __syncthreads();
#endif

    const int lane = threadIdx.x & 31;
    const int wave = threadIdx.x >> 5;
    const int wm = wave * 16;                 // wave's row offset in block tile
    if (m0 + wm < N) {                        // wave-uniform (rows are 16-multiples)
        const int half = lane >> 4;
        const int lid  = lane & 15;
        const int koff = half * 2;
        v8f acc = {0.f, 0.f, 0.f, 0.f, 0.f, 0.f, 0.f, 0.f};
        const float* arow = &As[(wm + lid) * K];
#pragma unroll
        for (int k = 0; k < K; k += 4) {
            v2f a, b;
            a[0] = arow[k + koff];
            a[1] = arow[k + koff + 1];
            b[0] = Bs[(k + koff) * 16 + lid];
            b[1] = Bs[(k + koff + 1) * 16 + lid];
            acc = __builtin_amdgcn_wmma_f32_16x16x4_f32(
                      false, a, false, b, (short)0, acc, false, false);
        }
        const int col = tn * 16 + lid;
        const int mbase = m0 + wm + half * 8;
#pragma unroll
        for (int r = 0; r < 8; ++r)
            out[(size_t)(mbase + r) * Cout + col] = acc[r];
    }
}

// ---------------------------------------------------------------------------
// GCN aggregation: agg[i] = hw[i]*dinv[i]^2 + bias  (self-loop term),
// then edge scatter agg[dst] += hw[src]*dinv[src]*dinv[dst].
// ---------------------------------------------------------------------------
__global__ void k_agg_init(const float4* __restrict__ hw,
                           const float* __restrict__ dinv,
                           const float4* __restrict__ bias,
                           float4* __restrict__ agg, int C4, int N) {
    int idx = blockIdx.x * blockDim.x + threadIdx.x;
    if (idx >= N * C4) return;
    int i = idx / C4, j = idx % C4;
    float w = dinv[i] * dinv[i];
    float4 h = hw[idx];
    float4 b = bias[j];
    float4 o;
    o.x = h.x * w + b.x; o.y = h.y * w + b.y;
    o.z = h.z * w + b.z; o.w = h.w * w + b.w;
    agg[idx] = o;
}

__global__ void k_agg_edges(const float4* __restrict__ hw,
                            const int* __restrict__ src,
                            const int* __restrict__ dst,
                            const float* __restrict__ dinv,
                            float* __restrict__ agg, int C4, long total) {
    long idx = (long)blockIdx.x * blockDim.x + threadIdx.x;
    if (idx >= total) return;
    int e = (int)(idx / C4);
    int j = (int)(idx % C4);
    int s = src[e], d = dst[e];
    float w = dinv[s] * dinv[d];
    float4 h = hw[(size_t)s * C4 + j];
    float* a = agg + ((size_t)d * C4 + j) * 4;
    atomicAdd(a + 0, h.x * w);
    atomicAdd(a + 1, h.y * w);
    atomicAdd(a + 2, h.z * w);
    atomicAdd(a + 3, h.w * w);
}

// ---------------------------------------------------------------------------
// GraphNorm: gsum/gvar are G x C accumulators (zeroed by hipMemsetAsync).
// ---------------------------------------------------------------------------
__global__ void k_gn_sum(const float* __restrict__ x, const int* __restrict__ batch,
                         float* gsum, int C, int N) {
    int idx = blockIdx.x * blockDim.x + threadIdx.x;
    if (idx >= N * C) return;
    int i = idx / C, c = idx % C;
    atomicAdd(&gsum[batch[i] * C + c], x[idx]);
}

__global__ void k_gn_var(const float* __restrict__ x, const int* __restrict__ batch,
                         const float* __restrict__ gsum, const float* __restrict__ counts,
                         const float* __restrict__ na, float* gvar, int C, int N) {
    int idx = blockIdx.x * blockDim.x + threadIdx.x;
    if (idx >= N * C) return;
    int i = idx / C, c = idx % C;
    int b = batch[i];
    float cnt = fmaxf(counts[b], 1.0f);
    float mean = gsum[b * C + c] / cnt;
    float hs = x[idx] - na[c] * mean;
    atomicAdd(&gvar[b * C + c], hs * hs);
}

__global__ void k_gn_apply(const float* __restrict__ x, float* __restrict__ y,
                           const int* __restrict__ batch,
                           const float* __restrict__ gsum, const float* __restrict__ gvar,
                           const float* __restrict__ counts,
                           const float* __restrict__ ng, const float* __restrict__ na,
                           const float* __restrict__ nb, int C, int N, int do_relu) {
    int idx = blockIdx.x * blockDim.x + threadIdx.x;
    if (idx >= N * C) return;
    int i = idx / C, c = idx % C;
    int b = batch[i];
    float cnt = fmaxf(counts[b], 1.0f);
    float mean = gsum[b * C + c] / cnt;
    float var  = gvar[b * C + c] / cnt;
    float hs = x[idx] - na[c] * mean;
    float v = ng[c] * hs * rsqrtf(var + GN_EPS) + nb[c];
    if (do_relu) v = fmaxf(v, 0.0f);
    y[idx] = v;
}

// feat += y ; cur = relu(feat)
__global__ void k_residual(const float* __restrict__ y, float* feat, float* cur, int n) {
    int i = blockIdx.x * blockDim.x + threadIdx.x;
    if (i >= n) return;
    float f = feat[i] + y[i];
    feat[i] = f;
    cur[i] = fmaxf(f, 0.0f);
}

// global max pool: h >= 0 after relu, so unsigned-bit max == float max
__global__ void k_gmax(const float* __restrict__ cur, const int* __restrict__ batch,
                       unsigned int* gb, int N) {
    int idx = blockIdx.x * blockDim.x + threadIdx.x;
    if (idx >= N * 32) return;
    int i = idx >> 5, c = idx & 31;
    atomicMax(&gb[batch[i] * 32 + c], __float_as_uint(cur[idx]));
}

// tiny row-major 32x32 matmul: out[r,c] = sum_k A[r,k]*W[k,c] + b[c]
__global__ void k_mm32(const float* __restrict__ A, const float* __restrict__ W,
                       const float* __restrict__ b, float* out, int rows) {
    int idx = blockIdx.x * blockDim.x + threadIdx.x;
    if (idx >= rows * 32) return;
    int r = idx >> 5, c = idx & 31;
    float s = b[c];
#pragma unroll
    for (int k = 0; k < 32; ++k) s += A[r * 32 + k] * W[k * 32 + c];
    out[idx] = s;
}

__global__ void k_scatter_out(const float* __restrict__ vo, const int* __restrict__ batch,
                              float* out, int N) {
    int idx = blockIdx.x * blockDim.x + threadIdx.x;
    if (idx >= N * 32) return;
    int i = idx >> 5, c = idx & 31;
    out[idx] = vo[batch[i] * 32 + c];
}

// ---------------------------------------------------------------------------
// Host orchestration
// ---------------------------------------------------------------------------
extern "C" void kernel_launch(void* const* d_in, const int* in_sizes, int n_in,
                              void* d_out, int out_size, void* d_ws, size_t ws_size,
                              hipStream_t stream) {
    const float* x       = (const float*)d_in[0];
    const int*   ei      = (const int*)  d_in[1];
    const int*   batch   = (const int*)  d_in[2];
    const float* conv0_W = (const float*)d_in[3];
    const float* conv0_b = (const float*)d_in[4];
    const float* ng0     = (const float*)d_in[5];
    const float* na0     = (const float*)d_in[6];
    const float* nb0     = (const float*)d_in[7];
    const float* W1      = (const float*)d_in[8];
    const float* b1      = (const float*)d_in[9];
    const float* ng1     = (const float*)d_in[10];
    const float* na1     = (const float*)d_in[11];
    const float* nb1     = (const float*)d_in[12];
    const float* W2      = (const float*)d_in[13];
    const float* b2      = (const float*)d_in[14];
    const float* ng2     = (const float*)d_in[15];
    const float* na2     = (const float*)d_in[16];
    const float* nb2     = (const float*)d_in[17];
    const float* Wv      = (const float*)d_in[22];
    const float* bv      = (const float*)d_in[23];
    const float* Wo      = (const float*)d_in[24];
    const float* bo      = (const float*)d_in[25];
    float* out = (float*)d_out;

    const int N = in_sizes[0] / 128;
    const int E = in_sizes[1] / 2;
    const int G = NG;
    const int* src = ei;
    const int* dst = ei + E;

    // workspace carve-up (all segment sizes are multiples of 4 floats -> 16B aligned)
    float* ws = (float*)d_ws;
    float* dinv   = ws; ws += N;                  // deg -> rsqrt in place
    float* counts = ws; ws += G;
    float* gsum   = ws; ws += (size_t)G * 128;
    float* gvar   = ws; ws += (size_t)G * 128;
    float* cur    = ws; ws += (size_t)N * 32;
    float* feat   = ws; ws += (size_t)N * 32;
    float* wide   = ws; ws += (size_t)N * 128;
    float* hwb    = ws; ws += (size_t)N * 128;    // gemm output
    float* aggb   = ws; ws += (size_t)N * 128;    // aggregation output
    float* gmaxb  = ws; ws += (size_t)G * 32;     // bits of nonneg floats
    float* vbuf   = ws; ws += (size_t)G * 32;
    float* vobuf  = ws; ws += (size_t)G * 32;

    const int T = 256;

    auto gemm = [&](const float* A, const float* B, float* O, int K, int Cout) {
        dim3 grid((N + 127) / 128, Cout / 16);
        if (K == 32)
            k_gemm_tdm<32, 128><<<grid, 256, 0, stream>>>(A, B, O, Cout, N);
        else
            k_gemm_tdm<128, 128><<<grid, 256, 0, stream>>>(A, B, O, Cout, N);
    };
    auto aggregate = [&](const float* hw, const float* bias, int C) {
        int C4 = C / 4;
        k_agg_init<<<(N * C4 + T - 1) / T, T, 0, stream>>>(
            (const float4*)hw, dinv, (const float4*)bias, (float4*)aggb, C4, N);
        long tot = (long)E * C4;
        k_agg_edges<<<(unsigned)((tot + T - 1) / T), T, 0, stream>>>(
            (const float4*)hw, src, dst, dinv, aggb, C4, tot);
    };
    auto gnorm = [&](const float* ng, const float* na, const float* nb,
                     float* dstp, int C) {
        hipMemsetAsync(gsum, 0, (size_t)G * C * sizeof(float), stream);
        hipMemsetAsync(gvar, 0, (size_t)G * C * sizeof(float), stream);
        int n = N * C;
        k_gn_sum<<<(n + T - 1) / T, T, 0, stream>>>(aggb, batch, gsum, C, N);
        k_gn_var<<<(n + T - 1) / T, T, 0, stream>>>(aggb, batch, gsum, counts, na, gvar, C, N);
        k_gn_apply<<<(n + T - 1) / T, T, 0, stream>>>(aggb, dstp, batch, gsum, gvar,
                                                      counts, ng, na, nb, C, N, 1);
    };

    // ---- degree / counts precompute ----
    k_fill<<<(N + T - 1) / T, T, 0, stream>>>(dinv, 1.0f, N);       // self-loop
    k_deg_edges<<<(E + T - 1) / T, T, 0, stream>>>(dst, dinv, E);
    k_rsqrt_ip<<<(N + T - 1) / T, T, 0, stream>>>(dinv, N);
    k_fill<<<(G + T - 1) / T, T, 0, stream>>>(counts, 0.0f, G);
    k_counts<<<(N + T - 1) / T, T, 0, stream>>>(batch, counts, N);

    // ---- conv0: GCN(128->32) + GN + relu -> cur ----
    gemm(x, conv0_W, hwb, 128, 32);
    aggregate(hwb, conv0_b, 32);
    gnorm(ng0, na0, nb0, cur, 32);

    // ---- 4 blocks x 3 residual layers ----
    for (int li = 0; li < 12; ++li) {
        if (li % 3 == 0)
            hipMemcpyAsync(feat, cur, (size_t)N * 32 * sizeof(float),
                           hipMemcpyDeviceToDevice, stream);

        // GCN(32->128) + GN + relu -> wide
        gemm(cur, W1 + (size_t)li * 32 * 128, hwb, 32, 128);
        aggregate(hwb, b1 + (size_t)li * 128, 128);
        gnorm(ng1 + (size_t)li * 128, na1 + (size_t)li * 128, nb1 + (size_t)li * 128,
              wide, 128);

        // GCN(128->32) + GN + relu -> y (reuse hwb region)
        gemm(wide, W2 + (size_t)li * 128 * 32, hwb, 128, 32);
        aggregate(hwb, b2 + (size_t)li * 32, 32);
        gnorm(ng2 + (size_t)li * 32, na2 + (size_t)li * 32, nb2 + (size_t)li * 32,
              hwb, 32);

        // feat += y ; cur = relu(feat)
        k_residual<<<(N * 32 + T - 1) / T, T, 0, stream>>>(hwb, feat, cur, N * 32);
    }

    // ---- global max pool + degenerate single-key MHA ----
    hipMemsetAsync(gmaxb, 0, (size_t)G * 32 * sizeof(float), stream);
    k_gmax<<<(N * 32 + T - 1) / T, T, 0, stream>>>(cur, batch, (unsigned int*)gmaxb, N);
    // v = gmax @ Wv + bv ; vo = v @ Wo + bo  (softmax over single key == 1)
    k_mm32<<<(G * 32 + T - 1) / T, T, 0, stream>>>(gmaxb, Wv, bv, vbuf, G);
    k_mm32<<<(G * 32 + T - 1) / T, T, 0, stream>>>(vbuf, Wo, bo, vobuf, G);
    k_scatter_out<<<(N * 32 + T - 1) / T, T, 0, stream>>>(vobuf, batch, out, N);
}